// Div_86887188398977
// MI455X (gfx1250) — compile-verified
//
#include <hip/hip_runtime.h>
#include <cstdint>

typedef float vf4 __attribute__((ext_vector_type(4)));

#define BLOCK_THREADS 256
#define TILE_H 8
#define TILE_W 1024
#define XSTR 1028   // floats per LDS x row: [3] = left halo, [4..1027] = 1024 data
#define YSTR 1024   // floats per LDS y row

constexpr int H = 2048;
constexpr int W = 2048;
constexpr int TILES_W = W / TILE_W;   // 2  (power of two)
constexpr int TILES_H = H / TILE_H;   // 256 (power of two)

// CDNA5 async global->LDS copy, tracked by ASYNCcnt (CDNA5 ISA).
__device__ __forceinline__ void async_ld_b128(uint32_t lds_byte_off, const void* gaddr) {
    asm volatile("global_load_async_to_lds_b128 %0, %1, off"
                 :: "v"(lds_byte_off), "v"(gaddr)
                 : "memory");
}

__device__ __forceinline__ void wait_asynccnt0() {
    asm volatile("s_wait_asynccnt 0" ::: "memory");
}

__global__ __launch_bounds__(BLOCK_THREADS)
void fd_sum_kernel(const float* __restrict__ x, const float* __restrict__ y,
                   const float* __restrict__ kx, const float* __restrict__ ky,
                   float* __restrict__ out)
{
    __shared__ float ldsX[TILE_H * XSTR];         // 32.9 KB
    __shared__ float ldsY[(TILE_H + 1) * YSTR];   // 36.9 KB

    const int t = threadIdx.x;                    // 0..255
    const unsigned blk = blockIdx.x;
    const int tw = (int)(blk & (TILES_W - 1));            // & 1
    const int th = (int)((blk >> 1) & (TILES_H - 1));     // >>1 & 255
    const int b  = (int)(blk >> 9);                       // image index

    const int w0 = tw * TILE_W;
    const int h0 = th * TILE_H;
    const size_t img = (size_t)b * (H * W);
    const bool topblk = (h0 == 0);
    const bool botblk = (h0 + TILE_H == H);

    // ---- Stage x tile rows: each thread async-copies one 16B chunk per row.
    const float* gx = x + img + (size_t)h0 * W + (w0 + t * 4);
#pragma unroll
    for (int r = 0; r < TILE_H; ++r) {
        uint32_t loff = (uint32_t)(uintptr_t)&ldsX[r * XSTR + 4 + t * 4];
        async_ld_b128(loff, gx + (size_t)r * W);
    }

    // ---- Stage y rows j=0..8 -> global row (h0-1+j). Row 0 is zero at the top
    // edge, row 8 is zero at the bottom edge (reference zero padding).
#pragma unroll
    for (int j = 0; j <= TILE_H; ++j) {
        const bool skip = (j == 0 && topblk) || (j == TILE_H && botblk);
        if (!skip) {
            const int gr = h0 - 1 + j;
            uint32_t loff = (uint32_t)(uintptr_t)&ldsY[j * YSTR + t * 4];
            async_ld_b128(loff, y + img + (size_t)gr * W + (w0 + t * 4));
        }
    }

    // ---- Boundary fills (regular DS stores) while async copies are in flight.
    vf4 z4 = {0.f, 0.f, 0.f, 0.f};
    if (topblk) *(vf4*)&ldsY[0 * YSTR + t * 4]      = z4;
    if (botblk) *(vf4*)&ldsY[TILE_H * YSTR + t * 4] = z4;
    if (t < TILE_H) {
        // left halo column: exactly the reference's zero pad when w0 == 0
        ldsX[t * XSTR + 3] = (w0 == 0)
            ? 0.0f
            : x[img + (size_t)(h0 + t) * W + (w0 - 1)];
    }

    wait_asynccnt0();      // per-wave: all async LDS writes landed
    __syncthreads();       // cross-wave visibility

    // ---- Compute 8 output rows per thread, one float4 column chunk each.
    const float kx0 = kx[0], kx1 = kx[1];
    const float ky0 = ky[0], ky1 = ky[1];
    const int  c = t * 4;
    const bool lastc = (w0 + c + 3 == W - 1);   // only lane 255 of the right tile
    float* go = out + img + (size_t)h0 * W + (w0 + c);

#pragma unroll
    for (int r = 0; r < TILE_H; ++r) {
        const float* xr = &ldsX[r * XSTR + 4 + c];
        vf4 xv = *(const vf4*)xr;                        // x[h, w..w+3]
        float xm = xr[-1];                               // x[h, w-1] (or 0 pad)
        vf4 yu = *(const vf4*)&ldsY[r * YSTR + c];       // y[h-1, w..w+3] (or 0)
        vf4 yc = *(const vf4*)&ldsY[(r + 1) * YSTR + c]; // y[h,   w..w+3] (or 0)
        float x3 = lastc ? 0.0f : xv.w;                  // x last column unused

        vf4 o;
        o.x = kx0 * xm   + kx1 * xv.x + ky0 * yu.x + ky1 * yc.x;
        o.y = kx0 * xv.x + kx1 * xv.y + ky0 * yu.y + ky1 * yc.y;
        o.z = kx0 * xv.y + kx1 * xv.z + ky0 * yu.z + ky1 * yc.z;
        o.w = kx0 * xv.z + kx1 * x3   + ky0 * yu.w + ky1 * yc.w;

        __builtin_nontemporal_store(o, (vf4*)(go + (size_t)r * W));
    }
}

extern "C" void kernel_launch(void* const* d_in, const int* in_sizes, int n_in,
                              void* d_out, int out_size, void* d_ws, size_t ws_size,
                              hipStream_t stream) {
    const float* x  = (const float*)d_in[0];
    const float* y  = (const float*)d_in[1];
    const float* kx = (const float*)d_in[2];
    const float* ky = (const float*)d_in[3];
    float* out = (float*)d_out;

    const int B = in_sizes[0] / (H * W);          // 16
    dim3 grid((unsigned)(B * TILES_H * TILES_W)); // 8192 blocks
    fd_sum_kernel<<<grid, BLOCK_THREADS, 0, stream>>>(x, y, kx, ky, out);
}